// GRUModel_7361573945900
// MI455X (gfx1250) — compile-verified
//
#include <hip/hip_runtime.h>
#include <math.h>

#define NN 100000
#define NE 1200000
#define CC 64
#define LAYERS 10
#define SCAN_BLOCKS ((NN + 255) / 256)  // 391

typedef __attribute__((ext_vector_type(2))) float v2f;
typedef __attribute__((ext_vector_type(8))) float v8f;

// Exact-fp32 WMMA: D(16x16,f32) = A(16x4,f32) * B(4x16,f32) + C
__device__ __forceinline__ v8f wmma4(v2f a, v2f b, v8f c) {
  return __builtin_amdgcn_wmma_f32_16x16x4_f32(false, a, false, b, (short)0, c,
                                               false, false);
}

__device__ __forceinline__ float sigmoid_(float x) {
  return 1.0f / (1.0f + __expf(-x));
}

// ===========================================================================
// CSR build (once per launch; edge_index is layer-invariant)
// ===========================================================================
__global__ __launch_bounds__(256) void zero_int_kernel(int* __restrict__ cnt,
                                                       int* __restrict__ cur) {
  const int i = blockIdx.x * 256 + threadIdx.x;
  if (i < NN) { cnt[i] = 0; cur[i] = 0; }
}

__global__ __launch_bounds__(256) void hist_kernel(const int* __restrict__ edge,
                                                   int* __restrict__ cnt) {
  const unsigned e = blockIdx.x * 256u + threadIdx.x;
  if (e < NE) atomicAdd(&cnt[edge[NE + e]], 1);
}

// Per-block exclusive scan (Hillis-Steele in LDS) + block totals.
__global__ __launch_bounds__(256) void scan1_kernel(const int* __restrict__ cnt,
                                                    int* __restrict__ row_start,
                                                    int* __restrict__ bsums) {
  __shared__ int tmp[256];
  const int tid = threadIdx.x;
  const int i = blockIdx.x * 256 + tid;
  const int v = (i < NN) ? cnt[i] : 0;
  int x = v;
  tmp[tid] = x;
  __syncthreads();
#pragma unroll
  for (int off = 1; off < 256; off <<= 1) {
    const int t = (tid >= off) ? tmp[tid - off] : 0;
    __syncthreads();
    x += t;
    tmp[tid] = x;
    __syncthreads();
  }
  if (i < NN) row_start[i] = x - v;      // block-local exclusive
  if (tid == 255) bsums[blockIdx.x] = x; // block total
}

__global__ void scan2_kernel(int* __restrict__ bsums) {
  if (blockIdx.x == 0 && threadIdx.x == 0) {
    int run = 0;
    for (int b = 0; b < SCAN_BLOCKS; ++b) {
      const int t = bsums[b];
      bsums[b] = run;
      run += t;
    }
  }
}

__global__ __launch_bounds__(256) void scan3_kernel(int* __restrict__ row_start,
                                                    const int* __restrict__ bsums) {
  const int i = blockIdx.x * 256 + threadIdx.x;
  if (i < NN) row_start[i] += bsums[blockIdx.x];
  if (i == 0) row_start[NN] = NE;  // total edge count is a compile-time constant
}

__global__ __launch_bounds__(256) void fill_kernel(const int* __restrict__ edge,
                                                   const int* __restrict__ row_start,
                                                   int* __restrict__ cursor,
                                                   int* __restrict__ csr_src) {
  const unsigned e = blockIdx.x * 256u + threadIdx.x;
  if (e < NE) {
    const int d = edge[NE + e];
    const int pos = row_start[d] + atomicAdd(&cursor[d], 1);
    csr_src[pos] = edge[e];
  }
}

// LDS row pad of 68 floats: v2f reads put lanes 0-15 on banks {4m,4m+1} and
// lanes 16-31 on {4m+2,4m+3} -> conflict-free across the 64 banks.
#define GPAD 68

// ===========================================================================
// Stage 1: msg = h @ W   (WMMA fp32). W is staged TRANSPOSED in LDS so the
// B-fragment is one contiguous ds_load_b64 (no v_mov repacking before WMMA).
// ===========================================================================
__global__ __launch_bounds__(256) void msg_gemm_kernel(
    const float* __restrict__ h, const float* __restrict__ W,
    float* __restrict__ msg) {
  extern __shared__ float sw[];  // 64 x GPAD floats, sw[n*GPAD + k] = W[k][n]
  for (int i = threadIdx.x; i < CC * CC; i += 256) {
    const int k = i >> 6, n = i & 63;  // coalesced global read of W[k][n]
    sw[n * GPAD + k] = W[i];
  }
  __syncthreads();

  const int wave = threadIdx.x >> 5;
  const int lane = threadIdx.x & 31;
  const int r0 = (blockIdx.x * 8 + wave) * 16;
  if (r0 >= NN) return;  // whole-wave exit: EXEC all-ones for WMMA

  const int ml = lane & 15;
  const int kh = (lane >> 4) << 1;

  v8f acc[4] = {};
  const float* arow = h + (size_t)(r0 + ml) * CC;

  for (int k = 0; k < CC; k += 4) {
    const int kk = k + kh;
    v2f a = *(const v2f*)(arow + kk);
#pragma unroll
    for (int t = 0; t < 4; ++t) {
      const int n = t * 16 + ml;
      v2f b = *(const v2f*)(sw + n * GPAD + kk);  // B[kk][n], B[kk+1][n]
      acc[t] = wmma4(a, b, acc[t]);
    }
  }

  const int hi8 = (lane >> 4) << 3;
#pragma unroll
  for (int t = 0; t < 4; ++t)
#pragma unroll
    for (int v = 0; v < 8; ++v)
      msg[(size_t)(r0 + hi8 + v) * CC + t * 16 + ml] = acc[t][v];
}

// ===========================================================================
// Stage 2: m[n] = sum over in-edges of msg[src]  (CSR gather, no atomics;
// msg is L2-resident so the random row gathers run at L2 speed)
// ===========================================================================
__global__ __launch_bounds__(256) void gather_sum_kernel(
    const float* __restrict__ msg, const int* __restrict__ row_start,
    const int* __restrict__ csr_src, float* __restrict__ m) {
  const unsigned tid = blockIdx.x * 256u + threadIdx.x;  // NN*16 threads exact
  const unsigned n = tid >> 4;
  const unsigned q = (tid & 15u) << 2;
  const int e0 = row_start[n];
  const int e1 = row_start[n + 1];
  float4 acc = make_float4(0.f, 0.f, 0.f, 0.f);
  for (int e = e0; e < e1; ++e) {
    const int s = csr_src[e];
    const float4 v = *(const float4*)(msg + (size_t)s * CC + q);
    acc.x += v.x;
    acc.y += v.y;
    acc.z += v.z;
    acc.w += v.w;
  }
  *(float4*)(m + (size_t)n * CC + q) = acc;
}

// ===========================================================================
// Stage 3: fused GRU cell via WMMA. w_ih/w_hh staged in LDS (w.T layout is
// already contiguous per B-fragment -> single ds_load_b64, conflict-free).
// ===========================================================================
__global__ __launch_bounds__(256) void gru_kernel(
    const float* __restrict__ m, const float* __restrict__ h,
    const float* __restrict__ w_ih, const float* __restrict__ w_hh,
    const float* __restrict__ b_ih, const float* __restrict__ b_hh,
    float* __restrict__ hout, int do_relu) {
  extern __shared__ float sw[];  // [192*GPAD w_ih][192*GPAD w_hh]
  float* swi = sw;
  float* swh = sw + 192 * GPAD;
  for (int i = threadIdx.x; i < 3 * CC * CC; i += 256) {
    const int r = i >> 6, c = i & 63;
    swi[r * GPAD + c] = w_ih[i];
    swh[r * GPAD + c] = w_hh[i];
  }
  __syncthreads();

  const int wave = threadIdx.x >> 5;
  const int lane = threadIdx.x & 31;
  const int r0 = (blockIdx.x * 8 + wave) * 16;
  if (r0 >= NN) return;

  const int ml = lane & 15;
  const int kh = (lane >> 4) << 1;

  v8f rz[8] = {};  // r/z gates: gi and gh accumulate into the same tiles
  v8f ia[4] = {};  // i_n
  v8f ha[4] = {};  // h_n

  const float* mrow = m + (size_t)(r0 + ml) * CC;
  const float* hrow = h + (size_t)(r0 + ml) * CC;

  for (int k = 0; k < CC; k += 4) {
    const int kk = k + kh;
    v2f am = *(const v2f*)(mrow + kk);
    v2f ah = *(const v2f*)(hrow + kk);
#pragma unroll
    for (int g = 0; g < 2; ++g) {
#pragma unroll
      for (int cb = 0; cb < 4; ++cb) {
        const int n = g * CC + cb * 16 + ml;
        v2f bi = *(const v2f*)(swi + n * GPAD + kk);
        v2f bh = *(const v2f*)(swh + n * GPAD + kk);
        rz[g * 4 + cb] = wmma4(am, bi, rz[g * 4 + cb]);
        rz[g * 4 + cb] = wmma4(ah, bh, rz[g * 4 + cb]);
      }
    }
#pragma unroll
    for (int cb = 0; cb < 4; ++cb) {
      const int n = 2 * CC + cb * 16 + ml;
      v2f bi = *(const v2f*)(swi + n * GPAD + kk);
      v2f bh = *(const v2f*)(swh + n * GPAD + kk);
      ia[cb] = wmma4(am, bi, ia[cb]);
      ha[cb] = wmma4(ah, bh, ha[cb]);
    }
  }

  const int hi8 = (lane >> 4) << 3;
#pragma unroll
  for (int cb = 0; cb < 4; ++cb) {
    const int c = cb * 16 + ml;
    const float br = b_ih[c] + b_hh[c];
    const float bz = b_ih[CC + c] + b_hh[CC + c];
    const float bin = b_ih[2 * CC + c];
    const float bhn = b_hh[2 * CC + c];
#pragma unroll
    for (int v = 0; v < 8; ++v) {
      const int row = r0 + hi8 + v;
      const float r = sigmoid_(rz[cb][v] + br);
      const float z = sigmoid_(rz[4 + cb][v] + bz);
      const float nn = tanhf(ia[cb][v] + bin + r * (ha[cb][v] + bhn));
      const float hv = h[(size_t)row * CC + c];
      float o = (1.0f - z) * nn + z * hv;
      if (do_relu) o = fmaxf(o, 0.0f);
      hout[(size_t)row * CC + c] = o;
    }
  }
}

// ===========================================================================
extern "C" void kernel_launch(void* const* d_in, const int* in_sizes, int n_in,
                              void* d_out, int out_size, void* d_ws,
                              size_t ws_size, hipStream_t stream) {
  const float* x = (const float*)d_in[0];
  const int* edge = (const int*)d_in[1];        // (2,E): src row, then dst row
  const float* weight = (const float*)d_in[2];  // [10,64,64]
  const float* w_ih = (const float*)d_in[3];    // [192,64]
  const float* w_hh = (const float*)d_in[4];    // [192,64]
  const float* b_ih = (const float*)d_in[5];    // [192]
  const float* b_hh = (const float*)d_in[6];    // [192]
  float* out = (float*)d_out;

  const size_t per = (size_t)NN * CC;  // 6.4M floats
  float* msg = (float*)d_ws;
  float* mbuf = msg + per;
  float* h_a = mbuf + per;
  float* h_b = h_a + per;
  int* cnt = (int*)(h_b + per);     // NN
  int* row_start = cnt + NN;        // NN+1
  int* cursor = row_start + NN + 1; // NN
  int* bsums = cursor + NN;         // SCAN_BLOCKS (padded to 512)
  int* csr_src = bsums + 512;       // NE

  const int gemm_blocks = (NN / 16 + 7) / 8;   // 782
  const int gath_blocks = (NN * 16) / 256;     // 6250 (exact)
  const int edge_blocks = (NE + 255) / 256;    // 4688

  // ---- CSR build (amortized over 10 layers) ----
  zero_int_kernel<<<SCAN_BLOCKS, 256, 0, stream>>>(cnt, cursor);
  hist_kernel<<<edge_blocks, 256, 0, stream>>>(edge, cnt);
  scan1_kernel<<<SCAN_BLOCKS, 256, 0, stream>>>(cnt, row_start, bsums);
  scan2_kernel<<<1, 1, 0, stream>>>(bsums);
  scan3_kernel<<<SCAN_BLOCKS, 256, 0, stream>>>(row_start, bsums);
  fill_kernel<<<edge_blocks, 256, 0, stream>>>(edge, row_start, cursor, csr_src);

  // ---- layer pipeline ----
  const size_t msg_lds = (size_t)CC * GPAD * sizeof(float);        // 17.4 KB
  const size_t gru_lds = (size_t)2 * 192 * GPAD * sizeof(float);   // 104.4 KB

  const float* h_in = x;
  for (int layer = 0; layer < LAYERS; ++layer) {
    float* h_next = (layer == LAYERS - 1) ? out : ((layer & 1) ? h_b : h_a);
    msg_gemm_kernel<<<gemm_blocks, 256, msg_lds, stream>>>(
        h_in, weight + (size_t)layer * CC * CC, msg);
    gather_sum_kernel<<<gath_blocks, 256, 0, stream>>>(msg, row_start, csr_src,
                                                       mbuf);
    gru_kernel<<<gemm_blocks, 256, gru_lds, stream>>>(
        mbuf, h_in, w_ih, w_hh, b_ih, b_hh, h_next,
        (layer == LAYERS - 1) ? 1 : 0);
    h_in = h_next;
  }
}